// Encoder_69870527971691
// MI455X (gfx1250) — compile-verified
//
#include <hip/hip_runtime.h>

typedef __attribute__((ext_vector_type(2))) float v2f;
typedef __attribute__((ext_vector_type(8))) float v8f;

#define NNODES 50000
#define NEDGES 800000

// ---------------------------------------------------------------------------
// Degree build: deg initialized to 1.0 (self loop), then +1 per incoming edge.
// ---------------------------------------------------------------------------
__global__ void k_deg_init(float* __restrict__ deg, int n) {
  int i = blockIdx.x * blockDim.x + threadIdx.x;
  if (i < n) deg[i] = 1.0f;
}

__global__ void k_deg_count(const int* __restrict__ dst, float* __restrict__ deg, int e) {
  int i = blockIdx.x * blockDim.x + threadIdx.x;
  if (i < e) unsafeAtomicAdd(&deg[dst[i]], 1.0f);
}

__global__ void k_dinv(float* __restrict__ deg, int n) {
  int i = blockIdx.x * blockDim.x + threadIdx.x;
  if (i < n) deg[i] = rsqrtf(deg[i]);   // deg >= 1 always (self loop)
}

// ---------------------------------------------------------------------------
// T = (RELU ? relu(X) : X) @ W   using V_WMMA_F32_16X16X4_F32.
//
// W (K x C) is staged into LDS once per block via the CDNA5 async copy path:
// global_load_async_to_lds_b128 (ASYNCcnt) + s_wait_asynccnt, then all 8
// waves read B fragments from LDS with immediate-offset ds_load_b32.
//
// K, C, RELU are compile-time: the K-loop fully unrolls; all A/B/D accesses
// use immediate offsets (no per-iteration address VALU).
//
// One wave -> one 16x16 output tile, K stepped by 4.
// A (16x4 f32): lanes 0-15 hold row M=lane, K=k..k+1; lanes 16-31 row
//               M=lane-16, K=k+2..k+3.
// B (4x16 f32): VGPR0 lanes0-15 = row k,   lanes16-31 = row k+2;
//               VGPR1 lanes0-15 = row k+1, lanes16-31 = row k+3.
// D (16x16 f32): VGPR v -> M = v + 8*(lane>=16), N = lane&15.
// ---------------------------------------------------------------------------
template <int K, int C, bool RELU>
__global__ void k_gemm_wmma(const float* __restrict__ X, const float* __restrict__ W,
                            float* __restrict__ T, int n) {
  __shared__ float sW[K * C];

  // --- async stage W -> LDS (16B per lane per iteration; exact fit) --------
  int tid = threadIdx.x;
#pragma unroll
  for (int i = 0; i < (K * C) / 4 / 256; ++i) {
    int idx = tid + i * 256;
    unsigned ldsb = (unsigned)(uintptr_t)(&sW[idx << 2]);  // low 32b = LDS byte offset
    const float* gp = W + ((size_t)idx << 2);
    asm volatile("global_load_async_to_lds_b128 %0, %1, off"
                 :: "v"(ldsb), "v"(gp)
                 : "memory");
  }
  asm volatile("s_wait_asynccnt 0x0" ::: "memory");
  __syncthreads();

  // --- per-wave 16x16 tile -------------------------------------------------
  int wave = (blockIdx.x * blockDim.x + threadIdx.x) >> 5;
  int lane = threadIdx.x & 31;
  constexpr int colTiles = C / 16;
  int rowTiles = n >> 4;
  if (wave >= rowTiles * colTiles) return;    // wave-uniform, after all barriers
  int rt = wave / colTiles;
  int ct = wave - rt * colTiles;
  int row0 = rt << 4;
  int col0 = ct << 4;
  int lh = lane >> 4;     // 0 or 1 (K-half)
  int lm = lane & 15;

  // Fold all runtime terms into base pointers; loop offsets are immediates.
  const float* ap = X + (size_t)(row0 + lm) * K + 2 * lh;     // 8B aligned
  const float* bq = sW + col0 + lm + 2 * lh * C;              // LDS base

  v8f acc = {};
#pragma unroll
  for (int k = 0; k < K; k += 4) {
    float2 av = *(const float2*)(ap + k);                     // global_load_b64 off:k*4
    if (RELU) { av.x = fmaxf(av.x, 0.0f); av.y = fmaxf(av.y, 0.0f); }
    v2f a, b;
    a.x = av.x;
    a.y = av.y;
    b.x = bq[k * C];                                          // ds_load_b32 off:imm
    b.y = bq[(k + 1) * C];                                    // ds_load_b32 off:imm
    acc = __builtin_amdgcn_wmma_f32_16x16x4_f32(
        /*neg_a=*/false, a, /*neg_b=*/false, b,
        /*c_mod=*/(short)0, acc, /*reuse_a=*/false, /*reuse_b=*/false);
  }

  float* tp = T + (size_t)row0 * C + col0 + lm + 8 * lh * C;
#pragma unroll
  for (int v = 0; v < 8; ++v)
    tp[v * C] = acc[v];                                       // store off:imm
}

// ---------------------------------------------------------------------------
// out[i, q*4..q*4+3] = bias[..] + T[i, ..] * dinv[i]^2  (self-loop + bias init)
// One thread per float4. Cq compile-time -> div/mod become shifts.
// ---------------------------------------------------------------------------
template <int Cq>
__global__ void k_init_out4(const float* __restrict__ T, const float* __restrict__ dinv,
                            const float* __restrict__ bias, float* __restrict__ out,
                            int n) {
  int i = blockIdx.x * blockDim.x + threadIdx.x;
  if (i >= n * Cq) return;
  int node = i / Cq;
  int q = i - node * Cq;
  float di = dinv[node];
  float s = di * di;
  float4 t = *(const float4*)(T + ((size_t)i << 2));
  float4 b = *(const float4*)(bias + (q << 2));
  float4 r;
  r.x = b.x + t.x * s;
  r.y = b.y + t.y * s;
  r.z = b.z + t.z * s;
  r.w = b.w + t.w * s;
  *(float4*)(out + ((size_t)i << 2)) = r;
}

// ---------------------------------------------------------------------------
// Edge scatter: out[dst, c] += T[src, c] * dinv[src] * dinv[dst]
// One thread per (edge, float4-of-channels): amortizes index/dinv loads 4x.
// Indices uniform across each 16 (C=64) / 8 (C=32) lane group -> scalarizable,
// L2-resident. Atomics lower to global_atomic_add_f32.
// ---------------------------------------------------------------------------
template <int C>
__global__ void k_scatter4(const int* __restrict__ src, const int* __restrict__ dst,
                           const float* __restrict__ T, const float* __restrict__ dinv,
                           float* __restrict__ out, int e) {
  constexpr int Q = C / 4;   // power of two
  long long tid = (long long)blockIdx.x * blockDim.x + threadIdx.x;
  long long eidx = tid / Q;
  int q = (int)(tid - eidx * Q);
  if (eidx >= e) return;
  int s = src[eidx];
  int d = dst[eidx];
  float nv = dinv[s] * dinv[d];
  float4 t = *(const float4*)(T + (size_t)s * C + (q << 2));
  float* op = out + (size_t)d * C + (q << 2);
  unsafeAtomicAdd(op + 0, t.x * nv);
  unsafeAtomicAdd(op + 1, t.y * nv);
  unsafeAtomicAdd(op + 2, t.z * nv);
  unsafeAtomicAdd(op + 3, t.w * nv);
}

// ---------------------------------------------------------------------------
// Host-side orchestration
// ---------------------------------------------------------------------------
extern "C" void kernel_launch(void* const* d_in, const int* in_sizes, int n_in,
                              void* d_out, int out_size, void* d_ws, size_t ws_size,
                              hipStream_t stream) {
  (void)in_sizes; (void)n_in; (void)out_size; (void)ws_size;

  const int N = NNODES;
  const int E = NEDGES;

  const float* x   = (const float*)d_in[0];
  const int*   ei  = (const int*)d_in[1];
  const int*   src = ei;          // edge_index[0]  (message sources)
  const int*   dst = ei + E;      // edge_index[1]  (aggregation targets)
  const float* W1  = (const float*)d_in[2];
  const float* b1  = (const float*)d_in[3];
  const float* W2  = (const float*)d_in[4];
  const float* b2  = (const float*)d_in[5];
  const float* Wmu = (const float*)d_in[6];
  const float* bmu = (const float*)d_in[7];
  const float* Wls = (const float*)d_in[8];
  const float* bls = (const float*)d_in[9];
  float* out = (float*)d_out;

  char* ws = (char*)d_ws;
  float* dinv = (float*)ws;
  size_t off = ((size_t)N * sizeof(float) + 255) & ~(size_t)255;
  float* bufA = (float*)(ws + off);                 // N x 64 f32 (pre-agg transforms)
  float* bufB = bufA + (size_t)N * 64;              // N x 64 f32 (aggregated, pre-relu)

  const int TB = 256;
  auto blks = [](long long work, int tb) { return (unsigned)((work + tb - 1) / tb); };
  auto gemm_blocks = [&](int n, int C) {
    long long waves = (long long)(n >> 4) * (C >> 4);
    return (unsigned)((waves + 7) / 8);             // 8 waves / 256-thread block
  };

  // --- degrees -> dinv (in place) ---
  k_deg_init<<<blks(N, TB), TB, 0, stream>>>(dinv, N);
  k_deg_count<<<blks(E, TB), TB, 0, stream>>>(dst, dinv, E);
  k_dinv<<<blks(N, TB), TB, 0, stream>>>(dinv, N);

  // --- layer 1: agg1 = scatter(x @ W1) + b1   (relu deferred to next load) ---
  k_gemm_wmma<128, 64, false><<<gemm_blocks(N, 64), TB, 0, stream>>>(x, W1, bufA, N);
  k_init_out4<16><<<blks((long long)N * 16, TB), TB, 0, stream>>>(bufA, dinv, b1, bufB, N);
  k_scatter4<64><<<blks((long long)E * 16, TB), TB, 0, stream>>>(src, dst, bufA, dinv, bufB, E);

  // --- layer 2: agg2 = scatter(relu(agg1) @ W2) + b2 ---
  k_gemm_wmma<64, 64, true><<<gemm_blocks(N, 64), TB, 0, stream>>>(bufB, W2, bufA, N);
  k_init_out4<16><<<blks((long long)N * 16, TB), TB, 0, stream>>>(bufA, dinv, b2, bufB, N);
  k_scatter4<64><<<blks((long long)E * 16, TB), TB, 0, stream>>>(src, dst, bufA, dinv, bufB, E);

  // --- layer 3: mu = agg(relu(agg2) @ Wmu) + bmu ; ls likewise ---
  float* tmu = bufA;                      // N x 32
  float* tls = bufA + (size_t)N * 32;     // N x 32
  float* omu = out;                       // N x 32
  float* ols = out + (size_t)N * 32;      // N x 32

  k_gemm_wmma<64, 32, true><<<gemm_blocks(N, 32), TB, 0, stream>>>(bufB, Wmu, tmu, N);
  k_gemm_wmma<64, 32, true><<<gemm_blocks(N, 32), TB, 0, stream>>>(bufB, Wls, tls, N);

  k_init_out4<8><<<blks((long long)N * 8, TB), TB, 0, stream>>>(tmu, dinv, bmu, omu, N);
  k_init_out4<8><<<blks((long long)N * 8, TB), TB, 0, stream>>>(tls, dinv, bls, ols, N);
  k_scatter4<32><<<blks((long long)E * 8, TB), TB, 0, stream>>>(src, dst, tmu, dinv, omu, E);
  k_scatter4<32><<<blks((long long)E * 8, TB), TB, 0, stream>>>(src, dst, tls, dinv, ols, E);
}